// GCN_75977971466925
// MI455X (gfx1250) — compile-verified
//
#include <hip/hip_runtime.h>
#include <math.h>

typedef float v2f __attribute__((ext_vector_type(2)));
typedef float v8f __attribute__((ext_vector_type(8)));

// ---------------------------------------------------------------------------
// Elementwise / CSR-building helpers
// ---------------------------------------------------------------------------
__global__ void zero_i_k(int* __restrict__ p, int n) {
    int t = blockIdx.x * blockDim.x + threadIdx.x;
    if (t < n) p[t] = 0;
}

__global__ void degree_i_k(const int* __restrict__ dst, int* __restrict__ degi, int E) {
    int t = blockIdx.x * blockDim.x + threadIdx.x;
    if (t < E) atomicAdd(&degi[dst[t]], 1);
}

__global__ void dinv_k(const int* __restrict__ degi, float* __restrict__ dinv, int n) {
    int t = blockIdx.x * blockDim.x + threadIdx.x;
    if (t < n) dinv[t] = rsqrtf((float)degi[t] + 1.0f);
}

// Single-workgroup exclusive scan of degi[0..n) -> rowptr[0..n]
__global__ __launch_bounds__(1024) void scan_k(const int* __restrict__ degi,
                                               int* __restrict__ rowptr, int n) {
    __shared__ int partial[1024];
    const int t = threadIdx.x;
    const int per = (n + 1023) / 1024;
    const int start = t * per;
    const int end = (start + per < n) ? (start + per) : n;
    int sum = 0;
    for (int i = start; i < end; ++i) sum += degi[i];
    partial[t] = sum;
    __syncthreads();
    // Hillis-Steele inclusive scan over 1024 partials
    for (int off = 1; off < 1024; off <<= 1) {
        int v = (t >= off) ? partial[t - off] : 0;
        __syncthreads();
        partial[t] += v;
        __syncthreads();
    }
    int run = (t == 0) ? 0 : partial[t - 1];  // exclusive prefix for this chunk
    for (int i = start; i < end; ++i) { rowptr[i] = run; run += degi[i]; }
    if (t == 1023) rowptr[n] = run;  // == E
}

__global__ void copy_i_k(int* __restrict__ dstp, const int* __restrict__ srcp, int n) {
    int t = blockIdx.x * blockDim.x + threadIdx.x;
    if (t < n) dstp[t] = srcp[t];
}

__global__ void scatter_k(const int* __restrict__ src, const int* __restrict__ dst,
                          int* __restrict__ cursor, int* __restrict__ csr_src, int E) {
    int t = blockIdx.x * blockDim.x + threadIdx.x;
    if (t < E) {
        int d = dst[t];
        int slot = atomicAdd(&cursor[d], 1);
        csr_src[slot] = src[t];
    }
}

// ---------------------------------------------------------------------------
// FP32 WMMA GEMM:  C[n x H] = (relu?)(X[n x K]) @ W[K x H]
// One wave per 16x16 output tile; H/16 waves per block; 16-row stripe per block.
// ---------------------------------------------------------------------------
template <int KT>
__global__ void gemm_wmma_f32(const float* __restrict__ X, const float* __restrict__ W,
                              float* __restrict__ C, int n, int K, int H, int doRelu) {
    extern __shared__ float lds[];
    float* lds_a = lds;                  // 16 x (KT+2)
    float* lds_b = lds + 16 * (KT + 2);  // KT x H

    const int t    = threadIdx.x;
    const int wave = t >> 5;
    const int lane = t & 31;
    const int half = lane >> 4;
    const int lm   = lane & 15;
    const int row0 = blockIdx.x * 16;
    const int n0   = wave * 16;

    v8f c = {};

    for (int k0 = 0; k0 < K; k0 += KT) {
        for (int idx = t; idx < 16 * KT; idx += blockDim.x) {
            int r  = idx / KT;
            int kk = idx % KT;
            int gr = row0 + r;
            int gk = k0 + kk;
            float v = (gr < n && gk < K) ? X[(size_t)gr * K + gk] : 0.0f;
            if (doRelu) v = fmaxf(v, 0.0f);
            lds_a[r * (KT + 2) + kk] = v;
        }
        for (int idx = t; idx < KT * H; idx += blockDim.x) {
            int kk  = idx / H;
            int col = idx % H;
            int gk  = k0 + kk;
            lds_b[kk * H + col] = (gk < K) ? W[(size_t)gk * H + col] : 0.0f;
        }
        __syncthreads();
#pragma unroll
        for (int kk = 0; kk < KT; kk += 4) {
            int ka = kk + 2 * half;
            v2f a, b;
            a.x = lds_a[lm * (KT + 2) + ka];
            a.y = lds_a[lm * (KT + 2) + ka + 1];
            b.x = lds_b[ka * H + n0 + lm];
            b.y = lds_b[(ka + 1) * H + n0 + lm];
            c = __builtin_amdgcn_wmma_f32_16x16x4_f32(false, a, false, b,
                                                      (short)0, c, false, false);
        }
        __syncthreads();
    }
#pragma unroll
    for (int vv = 0; vv < 8; ++vv) {
        int r = row0 + vv + 8 * half;
        if (r < n) C[(size_t)r * H + n0 + lm] = c[vv];
    }
}

// ---------------------------------------------------------------------------
// Pull aggregation (no atomics): one wave32 per destination node.
//   agg[i,:] = dinv[i]^2 * h[i,:] + bias + sum_{s in N(i)} dinv[s]*dinv[i]*h[s,:]
// ---------------------------------------------------------------------------
template <int H>
__global__ void pull_agg_k(const float* __restrict__ h, const float* __restrict__ dinv,
                           const float* __restrict__ bias, const int* __restrict__ rowptr,
                           const int* __restrict__ csr_src, float* __restrict__ agg, int n) {
    constexpr int VEC = H / 32;
    int i    = (blockIdx.x * blockDim.x + threadIdx.x) >> 5;
    int lane = threadIdx.x & 31;
    if (i >= n) return;
    const int f0 = lane * VEC;
    const float di = dinv[i];

    float acc[VEC];
    const float* hself = h + (size_t)i * H + f0;
#pragma unroll
    for (int v = 0; v < VEC; ++v) acc[v] = di * di * hself[v] + bias[f0 + v];

    const int jb = rowptr[i], je = rowptr[i + 1];
    for (int j = jb; j < je; ++j) {
        int s = csr_src[j];
        float norm = dinv[s] * di;
        const float* hp = h + (size_t)s * H + f0;
#pragma unroll
        for (int v = 0; v < VEC; ++v) acc[v] += hp[v] * norm;
    }

    float* ap = agg + (size_t)i * H + f0;
#pragma unroll
    for (int v = 0; v < VEC; ++v) ap[v] = acc[v];
}

// ---------------------------------------------------------------------------
// Output head: out[i, 0..1] = relu(agg[i, 0..63]) @ Wo[64x2] + bo
// ---------------------------------------------------------------------------
__global__ void out_head_k(const float* __restrict__ agg, const float* __restrict__ Wo,
                           const float* __restrict__ bo, float* __restrict__ out, int n) {
    int i = blockIdx.x * blockDim.x + threadIdx.x;
    if (i >= n) return;
    float acc0 = bo[0], acc1 = bo[1];
    const float4* ap = (const float4*)(agg + (size_t)i * 64);
#pragma unroll
    for (int f4 = 0; f4 < 16; ++f4) {
        float4 v = ap[f4];
        float v0 = fmaxf(v.x, 0.0f), v1 = fmaxf(v.y, 0.0f);
        float v2 = fmaxf(v.z, 0.0f), v3 = fmaxf(v.w, 0.0f);
        int f = f4 * 4;
        acc0 += v0 * Wo[(f + 0) * 2 + 0] + v1 * Wo[(f + 1) * 2 + 0] +
                v2 * Wo[(f + 2) * 2 + 0] + v3 * Wo[(f + 3) * 2 + 0];
        acc1 += v0 * Wo[(f + 0) * 2 + 1] + v1 * Wo[(f + 1) * 2 + 1] +
                v2 * Wo[(f + 2) * 2 + 1] + v3 * Wo[(f + 3) * 2 + 1];
    }
    out[(size_t)i * 2 + 0] = acc0;
    out[(size_t)i * 2 + 1] = acc1;
}

// ---------------------------------------------------------------------------
static inline size_t align_up(size_t x, size_t a) { return (x + a - 1) & ~(a - 1); }

extern "C" void kernel_launch(void* const* d_in, const int* in_sizes, int n_in,
                              void* d_out, int out_size, void* d_ws, size_t ws_size,
                              hipStream_t stream) {
    const float* x  = (const float*)d_in[0];
    const int*   ei = (const int*)d_in[1];
    const float* W1 = (const float*)d_in[2];
    const float* b1 = (const float*)d_in[3];
    const float* W2 = (const float*)d_in[4];
    const float* b2 = (const float*)d_in[5];
    const float* Wo = (const float*)d_in[6];
    const float* bo = (const float*)d_in[7];
    float* out = (float*)d_out;

    const int K1 = 165, H1 = 128, H2 = 64;
    const int n = in_sizes[0] / K1;
    const int E = in_sizes[1] / 2;
    const int* src = ei;
    const int* dst = ei + E;

    // ---- workspace layout (16B aligned regions) ----
    char* ws = (char*)d_ws;
    size_t off = 0;
    float* dinv   = (float*)(ws + off); off = align_up(off + (size_t)n * 4, 16);
    int*   degi   = (int*)(ws + off);   off = align_up(off + (size_t)n * 4, 16);
    int*   rowptr = (int*)(ws + off);   off = align_up(off + ((size_t)n + 1) * 4, 16);
    int*   cursor = (int*)(ws + off);   off = align_up(off + (size_t)n * 4, 16);
    int*   csr    = (int*)(ws + off);   off = align_up(off + (size_t)E * 4, 16);
    float* bufH   = (float*)(ws + off); off += (size_t)n * H1 * 4;
    float* bufA   = (float*)(ws + off);

    // ---- degrees, dinv, CSR (by dst) ----
    zero_i_k<<<(n + 255) / 256, 256, 0, stream>>>(degi, n);
    degree_i_k<<<(E + 255) / 256, 256, 0, stream>>>(dst, degi, E);
    dinv_k<<<(n + 255) / 256, 256, 0, stream>>>(degi, dinv, n);
    scan_k<<<1, 1024, 0, stream>>>(degi, rowptr, n);
    copy_i_k<<<(n + 255) / 256, 256, 0, stream>>>(cursor, rowptr, n);
    scatter_k<<<(E + 255) / 256, 256, 0, stream>>>(src, dst, cursor, csr, E);

    const int rowBlocks = (n + 15) / 16;

    // ---- layer 1: h1 = x @ W1 ; agg1 = pull(h1) + self-loop + b1 ----
    {
        size_t lds = (size_t)(16 * (32 + 2) + 32 * H1) * sizeof(float);
        gemm_wmma_f32<32><<<rowBlocks, 32 * (H1 / 16), lds, stream>>>(x, W1, bufH, n, K1, H1, 0);
    }
    pull_agg_k<128><<<(n + 7) / 8, 256, 0, stream>>>(bufH, dinv, b1, rowptr, csr, bufA, n);

    // ---- layer 2: h2 = relu(agg1) @ W2 ; agg2 = pull(h2) + self-loop + b2 ----
    {
        size_t lds = (size_t)(16 * (32 + 2) + 32 * H2) * sizeof(float);
        gemm_wmma_f32<32><<<rowBlocks, 32 * (H2 / 16), lds, stream>>>(bufA, W2, bufH, n, H1, H2, 1);
    }
    pull_agg_k<64><<<(n + 7) / 8, 256, 0, stream>>>(bufH, dinv, b2, rowptr, csr, bufA, n);

    // ---- output head ----
    out_head_k<<<(n + 255) / 256, 256, 0, stream>>>(bufA, Wo, bo, out, n);
}